// GNNRecommender_82188494176298
// MI455X (gfx1250) — compile-verified
//
#include <hip/hip_runtime.h>
#include <hip/hip_bf16.h>

typedef float v2f __attribute__((ext_vector_type(2)));
typedef float v8f __attribute__((ext_vector_type(8)));

#define NODES_F 256   // F_IN
#define HIDC 64       // per-head channels

// ---------------------------------------------------------------------------
// Pack B[K x NC] into pair-interleaved Bp[K/2][NC][2]:
//   Bp[((k>>1)*NC + c)*2 + (k&1)] = B[k*NC + c]
// Makes the per-lane WMMA B fragment {B[k][c], B[k+1][c]} contiguous (8B).
// ---------------------------------------------------------------------------
__global__ void gat_pack_pairs(const float* __restrict__ B,
                               float* __restrict__ Bp, int K, int NC) {
    int t = blockIdx.x * blockDim.x + threadIdx.x;
    if (t >= K * NC) return;
    int k = t / NC, c = t % NC;
    Bp[((size_t)(k >> 1) * NC + c) * 2 + (k & 1)] = B[t];
}

// ---------------------------------------------------------------------------
// fp32 WMMA GEMM: C[M x NC] = A[M x K] @ B[K x NC] (+ bias[NC])
// B pre-packed (pair-interleaved). Block = 8 waves, one 64-col group,
// 128 rows. The block's K x 64 packed panel is staged in LDS once
// (float4 fills), then each wave's inner loop (uniform trip count) is:
//   1x global_load_b64 (A) + 2x ds_load_2addr_b64 (B) + 4x v_wmma_f32_16x16x4_f32.
// Requires M % 16 == 0, K % 4 == 0, NC % 64 == 0 (true for all call sites).
// Dynamic LDS: K * 64 floats (= K*256 bytes).
// ---------------------------------------------------------------------------
__global__ void gat_gemm_wmma_f32(const float* __restrict__ A,
                                  const float* __restrict__ Bp,
                                  const float* __restrict__ bias,
                                  float* __restrict__ C,
                                  int M, int K, int NC) {
    extern __shared__ float panel[];               // [K/2][64][2] = K*64 floats

    const int colGroups = NC >> 6;
    const int rowTiles  = M >> 4;
    const int cg     = blockIdx.x % colGroups;
    const int rtBase = (blockIdx.x / colGroups) * 8;

    // ---- cooperative panel fill: (K/2) rows of 128 contiguous floats ----
    {
        const int units = K * 16;                  // float4 units in panel
        for (int u = threadIdx.x; u < units; u += blockDim.x) {
            int k2  = u >> 5;                      // 32 float4 per packed row
            int off = (u & 31) * 4;
            const float4 v = *(const float4*)(Bp + (size_t)k2 * (NC * 2) + cg * 128 + off);
            *(float4*)(panel + (size_t)k2 * 128 + off) = v;
        }
    }
    __syncthreads();

    const int lane = threadIdx.x & 31;
    const int rt   = rtBase + (threadIdx.x >> 5);
    if (rt >= rowTiles) return;

    const int hl  = lane >> 4;                     // 0: K pair {0,1}, 1: {2,3}
    const int l16 = lane & 15;
    const int arow = rt * 16 + l16;
    const int col0 = cg * 64 + l16;

    v8f acc0 = {}, acc1 = {}, acc2 = {}, acc3 = {};

    // lane-dependent start folded into base pointers -> uniform trip count
    const float* aptr = A + (size_t)arow * K + 2 * hl;
    const float* lptr = panel + hl * 128 + l16 * 2;
    const int iters = K >> 2;
    #pragma unroll 2
    for (int it = 0; it < iters; ++it) {
        v2f a  = *(const v2f*)aptr;  aptr += 4;
        v2f b0 = *(const v2f*)(lptr);
        v2f b1 = *(const v2f*)(lptr + 32);
        v2f b2 = *(const v2f*)(lptr + 64);
        v2f b3 = *(const v2f*)(lptr + 96);
        lptr += 256;                               // advance 2 packed K-rows
        acc0 = __builtin_amdgcn_wmma_f32_16x16x4_f32(false, a, false, b0, (short)0, acc0, false, false);
        acc1 = __builtin_amdgcn_wmma_f32_16x16x4_f32(false, a, false, b1, (short)0, acc1, false, false);
        acc2 = __builtin_amdgcn_wmma_f32_16x16x4_f32(false, a, false, b2, (short)0, acc2, false, false);
        acc3 = __builtin_amdgcn_wmma_f32_16x16x4_f32(false, a, false, b3, (short)0, acc3, false, false);
    }

    float bv0 = 0.f, bv1 = 0.f, bv2 = 0.f, bv3 = 0.f;
    if (bias) {
        bv0 = bias[col0];
        bv1 = bias[col0 + 16];
        bv2 = bias[col0 + 32];
        bv3 = bias[col0 + 48];
    }
    const int rbase = rt * 16 + 8 * hl;            // D vgpr i -> row rbase+i
    #pragma unroll
    for (int i = 0; i < 8; ++i) {
        float* p = C + (size_t)(rbase + i) * NC + col0;
        p[0]  = acc0[i] + bv0;
        p[16] = acc1[i] + bv1;
        p[32] = acc2[i] + bv2;
        p[48] = acc3[i] + bv3;
    }
}

// ---------------------------------------------------------------------------
// Per-(node, head): attention dot products al_s/al_d, init m=-inf, z=0, agg=0
// ---------------------------------------------------------------------------
__global__ void gat_node_prep(const float* __restrict__ h,
                              const float* __restrict__ a_src,
                              const float* __restrict__ a_dst,
                              float* __restrict__ al_s, float* __restrict__ al_d,
                              float* __restrict__ m, float* __restrict__ z,
                              float* __restrict__ agg,
                              int Nn, int H) {
    int t = blockIdx.x * blockDim.x + threadIdx.x;
    if (t >= Nn * H) return;
    int n = t / H, hd = t % H;
    const float* row = h + ((size_t)n * H + hd) * HIDC;
    const float* as  = a_src + hd * HIDC;
    const float* ad  = a_dst + hd * HIDC;
    float s = 0.f, d = 0.f;
    #pragma unroll 8
    for (int c = 0; c < HIDC; ++c) {
        float v = row[c];
        s = fmaf(v, as[c], s);
        d = fmaf(v, ad[c], d);
    }
    al_s[t] = s;
    al_d[t] = d;
    m[t] = -__builtin_inff();
    z[t] = 0.f;
    float* ag = agg + ((size_t)n * H + hd) * HIDC;
    #pragma unroll 8
    for (int c = 0; c < HIDC; ++c) ag[c] = 0.f;
}

__device__ __forceinline__ void edge_pair(const long long* __restrict__ ei,
                                          int e, int E, int& src, int& dst) {
    if (e < E) {
        src = (int)ei[e];
        dst = (int)ei[(size_t)E + e];
    } else {            // appended self-loop
        src = e - E;
        dst = e - E;
    }
}

__device__ __forceinline__ float leaky02(float x) {
    return x > 0.f ? x : 0.2f * x;
}

// monotone float atomic max via native integer atomics
__device__ __forceinline__ void atomicMaxF32(float* a, float v) {
    if (v >= 0.f) atomicMax((int*)a, __float_as_int(v));
    else          atomicMin((unsigned int*)a, __float_as_uint(v));
}

// ---------------------------------------------------------------------------
// Pass 1 over edges: per-dst running max of leaky_relu logits
// ---------------------------------------------------------------------------
__global__ void gat_edge_max(const long long* __restrict__ ei, int E, int Nn,
                             const float* __restrict__ al_s,
                             const float* __restrict__ al_d,
                             float* __restrict__ m, int H) {
    int t = blockIdx.x * blockDim.x + threadIdx.x;
    int total = (E + Nn) * H;
    if (t >= total) return;
    int e = t / H, hd = t % H;
    int src, dst;
    edge_pair(ei, e, E, src, dst);
    float logit = leaky02(al_s[src * H + hd] + al_d[dst * H + hd]);
    atomicMaxF32(&m[dst * H + hd], logit);
}

// ---------------------------------------------------------------------------
// Pass 2 over edges (fused): z[dst] += e ; agg[dst] += e * h[src]
// One thread per (edge, channel 0..63), loops over heads.
// ---------------------------------------------------------------------------
__global__ void gat_edge_accum(const long long* __restrict__ ei, int E, int Nn,
                               const float* __restrict__ al_s,
                               const float* __restrict__ al_d,
                               const float* __restrict__ m,
                               const float* __restrict__ h,
                               float* __restrict__ z,
                               float* __restrict__ agg, int H) {
    int t = blockIdx.x * blockDim.x + threadIdx.x;
    int total = (E + Nn) * HIDC;
    if (t >= total) return;
    int e = t >> 6;
    int c = t & (HIDC - 1);
    int src, dst;
    edge_pair(ei, e, E, src, dst);
    for (int hd = 0; hd < H; ++hd) {
        float logit = leaky02(al_s[src * H + hd] + al_d[dst * H + hd]);
        float ex = __expf(logit - m[dst * H + hd]);
        if (c == 0) atomicAdd(&z[dst * H + hd], ex);
        float msg = ex * h[((size_t)src * H + hd) * HIDC + c];
        atomicAdd(&agg[((size_t)dst * H + hd) * HIDC + c], msg);
    }
}

// ---------------------------------------------------------------------------
// out[n,h,c] = (opt relu)( agg[n,h,c] / z[n,h] + bias[h*64+c] )
// ---------------------------------------------------------------------------
__global__ void gat_node_finalize(const float* __restrict__ agg,
                                  const float* __restrict__ z,
                                  const float* __restrict__ bias,
                                  float* __restrict__ out,
                                  int Nn, int H, int doRelu) {
    int t = blockIdx.x * blockDim.x + threadIdx.x;
    int total = Nn * H * HIDC;
    if (t >= total) return;
    int hc = t % (H * HIDC);
    int hd = hc / HIDC;
    int n  = t / (H * HIDC);
    float v = agg[t] / z[n * H + hd] + bias[hc];
    if (doRelu) v = fmaxf(v, 0.f);
    out[t] = v;
}

// ---------------------------------------------------------------------------
extern "C" void kernel_launch(void* const* d_in, const int* in_sizes, int n_in,
                              void* d_out, int out_size, void* d_ws, size_t ws_size,
                              hipStream_t stream) {
    const float*     x   = (const float*)d_in[0];
    const long long* ei  = (const long long*)d_in[1];
    const float*     W1  = (const float*)d_in[2];
    const float*     as1 = (const float*)d_in[3];
    const float*     ad1 = (const float*)d_in[4];
    const float*     b1  = (const float*)d_in[5];
    const float*     W2  = (const float*)d_in[6];
    const float*     as2 = (const float*)d_in[7];
    const float*     ad2 = (const float*)d_in[8];
    const float*     b2  = (const float*)d_in[9];
    const float*     Wf  = (const float*)d_in[10];
    const float*     bfb = (const float*)d_in[11];
    float*           out = (float*)d_out;

    const int Nn = in_sizes[0] / NODES_F;   // 50000
    const int E  = in_sizes[1] / 2;         // 800000
    const int ET = E + Nn;                  // with self-loops

    float* ws  = (float*)d_ws;
    float* h1  = ws;                              // Nn*256
    float* x2  = ws + (size_t)Nn * 256;           // Nn*256 (agg1 -> relu'd input)
    float* st  = ws + (size_t)Nn * 512;
    float* als1 = st;                             // Nn*4
    float* ald1 = st + (size_t)Nn * 4;
    float* m1   = st + (size_t)Nn * 8;
    float* z1   = st + (size_t)Nn * 12;
    float* als2 = st + (size_t)Nn * 16;           // Nn*1 each
    float* ald2 = st + (size_t)Nn * 17;
    float* m2   = st + (size_t)Nn * 18;
    float* z2   = st + (size_t)Nn * 19;
    float* w1p  = st + (size_t)Nn * 20;           // packed weights
    float* w2p  = w1p + 256 * 256;
    float* wfp  = w2p + 256 * 64;
    float* h2   = h1;                             // reuse: Nn*64
    float* x3   = h1 + (size_t)Nn * 64;           // agg2 -> x3, Nn*64

    const int BLK = 256;
    auto gemm_grid = [](int M, int NC) {
        int rowBlocks = ((M >> 4) + 7) / 8;       // 8 row-tiles per block
        return rowBlocks * (NC >> 6);
    };

    // ---- pack weights (pair-interleaved for 8B B-fragment loads) ----
    gat_pack_pairs<<<(256 * 256 + BLK - 1) / BLK, BLK, 0, stream>>>(W1, w1p, 256, 256);
    gat_pack_pairs<<<(256 * 64  + BLK - 1) / BLK, BLK, 0, stream>>>(W2, w2p, 256, 64);
    gat_pack_pairs<<<(64  * 256 + BLK - 1) / BLK, BLK, 0, stream>>>(Wf, wfp, 64, 256);

    // ---- Layer 1 ----
    gat_gemm_wmma_f32<<<gemm_grid(Nn, 256), BLK, 256 * 256, stream>>>(x, w1p, nullptr, h1, Nn, 256, 256);
    gat_node_prep<<<(Nn * 4 + BLK - 1) / BLK, BLK, 0, stream>>>(h1, as1, ad1, als1, ald1, m1, z1, x2, Nn, 4);
    gat_edge_max<<<(ET * 4 + BLK - 1) / BLK, BLK, 0, stream>>>(ei, E, Nn, als1, ald1, m1, 4);
    gat_edge_accum<<<(ET * HIDC + BLK - 1) / BLK, BLK, 0, stream>>>(ei, E, Nn, als1, ald1, m1, h1, z1, x2, 4);
    gat_node_finalize<<<(Nn * 256 + BLK - 1) / BLK, BLK, 0, stream>>>(x2, z1, b1, x2, Nn, 4, 1);

    // ---- Layer 2 ---- (h2 overwrites h1 region; x2 is separate)
    gat_gemm_wmma_f32<<<gemm_grid(Nn, 64), BLK, 256 * 256, stream>>>(x2, w2p, nullptr, h2, Nn, 256, 64);
    gat_node_prep<<<(Nn + BLK - 1) / BLK, BLK, 0, stream>>>(h2, as2, ad2, als2, ald2, m2, z2, x3, Nn, 1);
    gat_edge_max<<<(ET + BLK - 1) / BLK, BLK, 0, stream>>>(ei, E, Nn, als2, ald2, m2, 1);
    gat_edge_accum<<<(ET * HIDC + BLK - 1) / BLK, BLK, 0, stream>>>(ei, E, Nn, als2, ald2, m2, h2, z2, x3, 1);
    gat_node_finalize<<<(Nn * 64 + BLK - 1) / BLK, BLK, 0, stream>>>(x3, z2, b2, x3, Nn, 1, 0);

    // ---- Final projection: out = x3 @ Wf + bf ----
    gat_gemm_wmma_f32<<<gemm_grid(Nn, 256), BLK, 64 * 256, stream>>>(x3, wfp, bfb, out, Nn, 64, 256);
}